// S6Block_64381559767209
// MI455X (gfx1250) — compile-verified
//
#include <hip/hip_runtime.h>
#include <math.h>

#define D_MODEL 512
#define N_STATE 64
#define SEQ_L   1024
#define NC      4
#define BATCH   8
#define HIDDEN  1024
#define EPS     1e-5f
#define KBIG    16384   /* NC * 64 * 64 */

typedef __attribute__((ext_vector_type(2))) float v2f;
typedef __attribute__((ext_vector_type(8))) float v8f;

#define WMMA_F32(A, B, C) \
  __builtin_amdgcn_wmma_f32_16x16x4_f32(false, (A), false, (B), (short)0, (C), false, false)

/* ---------------- workspace layout (float offsets) ---------------- */
#define WS_KREVT 0                                           /* 1024*512  */
#define WS_ABAR  (WS_KREVT + SEQ_L*D_MODEL)                  /* 4*64*64   */
#define WS_BBAR  (WS_ABAR + NC*N_STATE*N_STATE)              /* 4*512*64  */
#define WS_SMEAN (WS_BBAR + NC*D_MODEL*N_STATE)              /* 8*512     */
#define WS_SGATE (WS_SMEAN + BATCH*D_MODEL)                  /* 8*512     */
#define WS_BASE1 (WS_SGATE + BATCH*D_MODEL)
/* phase 1 (dead after k_gemm_K) */
#define WS_ST    WS_BASE1                                    /* 4*64*4096  */
#define WS_WBIG  (WS_ST + NC*64*N_STATE*N_STATE)             /* 512*16384  */
#define WS_QBIG  (WS_WBIG + (size_t)D_MODEL*KBIG)            /* 1024*16384 */
/* phase 2 (overlaps phase 1) */
#define WS_Y     WS_BASE1                                    /* 8*1024*512  */
#define WS_Y1    (WS_Y  + (size_t)BATCH*SEQ_L*D_MODEL)       /* 8*1024*512  */
#define WS_H     (WS_Y1 + (size_t)BATCH*SEQ_L*D_MODEL)       /* 8192*1024   */
#define WS_ZR    (WS_H  + (size_t)BATCH*SEQ_L*HIDDEN)        /* 8*1024*512  */

/* ============ K1: Cayley discretization via forward substitution ============ */
__global__ void k_discretize(const float* __restrict__ Bp,
                             const float* __restrict__ ldt_arr,
                             float* __restrict__ ws) {
  int c = blockIdx.x;
  int part = blockIdx.y;          /* 0: A_bar, 1..8: B_bar column groups */
  int tid = threadIdx.x;          /* 64 */
  float ldt = ldt_arr[c];
  float dt = (ldt > 20.f) ? ldt : log1pf(expf(ldt));
  dt += 1e-6f;
  float hdt = 0.5f * dt;
  if (part == 0) {
    int col = tid;
    float sqc = sqrtf(2.f * col + 1.f);
    float acc = 0.f;
    float* Ab = ws + WS_ABAR + c * 4096;
    for (int i = 0; i < 64; ++i) {
      float sqi = sqrtf(2.f * i + 1.f);
      float r;
      if (i == col)      r = 1.f - hdt * (i + 1.f);
      else if (i > col)  r = -hdt * sqi * sqc;
      else               r = 0.f;
      float x = (r - hdt * sqi * acc) / (1.f + hdt * (i + 1.f));
      Ab[i * 64 + col] = x;
      acc += sqi * x;
    }
  } else {
    int dd = (part - 1) * 64 + tid;
    const float* Brow = Bp + ((size_t)c * D_MODEL + dd) * N_STATE;
    float* Bb = ws + WS_BBAR + ((size_t)c * D_MODEL + dd) * N_STATE;
    float acc = 0.f;
    for (int i = 0; i < 64; ++i) {
      float sqi = sqrtf(2.f * i + 1.f);
      float x = (dt * Brow[i] - hdt * sqi * acc) / (1.f + hdt * (i + 1.f));
      Bb[i] = x;
      acc += sqi * x;
    }
  }
}

/* ============ K2: serial base powers S_j = A^j (j<32), T_k = A^(32k) ======= */
__global__ void k_powers(float* __restrict__ ws) {
  int c = blockIdx.x, tid = threadIdx.x; /* 256 */
  __shared__ float aM[4096], b0[4096], b1[4096];
  float* st = ws + WS_ST + (size_t)c * 64 * 4096;
  const float* Ab = ws + WS_ABAR + c * 4096;
  for (int e = tid; e < 4096; e += 256) aM[e] = Ab[e];
  float* cur = b0; float* nxt = b1;
  for (int e = tid; e < 4096; e += 256) {
    float id = ((e >> 6) == (e & 63)) ? 1.f : 0.f;
    cur[e] = id;
    st[e] = id;                 /* S_0 = I */
    st[32 * 4096 + e] = id;     /* T_0 = I */
  }
  __syncthreads();
  for (int j = 1; j < 32; ++j) {
    for (int e = tid; e < 4096; e += 256) {
      int i = e >> 6, jj = e & 63; float s = 0.f;
      for (int m = 0; m < 64; ++m) s += cur[i * 64 + m] * aM[m * 64 + jj];
      nxt[e] = s;
    }
    __syncthreads();
    for (int e = tid; e < 4096; e += 256) st[j * 4096 + e] = nxt[e];
    float* t = cur; cur = nxt; nxt = t;
    __syncthreads();
  }
  /* cur = S_31 ; A32 = S_31 @ A = T_1 */
  for (int e = tid; e < 4096; e += 256) {
    int i = e >> 6, jj = e & 63; float s = 0.f;
    for (int m = 0; m < 64; ++m) s += cur[i * 64 + m] * aM[m * 64 + jj];
    nxt[e] = s;
  }
  __syncthreads();
  for (int e = tid; e < 4096; e += 256) st[33 * 4096 + e] = nxt[e];
  __syncthreads();
  for (int e = tid; e < 4096; e += 256) aM[e] = nxt[e];   /* multiplier = A^32 */
  { float* t = cur; cur = nxt; nxt = t; }
  __syncthreads();
  for (int k = 2; k < 32; ++k) {
    for (int e = tid; e < 4096; e += 256) {
      int i = e >> 6, jj = e & 63; float s = 0.f;
      for (int m = 0; m < 64; ++m) s += cur[i * 64 + m] * aM[m * 64 + jj];
      nxt[e] = s;
    }
    __syncthreads();
    for (int e = tid; e < 4096; e += 256) st[(32 + k) * 4096 + e] = nxt[e];
    float* t = cur; cur = nxt; nxt = t;
    __syncthreads();
  }
}

/* ============ K3: Q[32k+j] = T_k @ S_j (all 4096 matmuls parallel) ========= */
__global__ void k_buildQ(float* __restrict__ ws) {
  int jj = blockIdx.x, kk = blockIdx.y, c = blockIdx.z;
  int tid = threadIdx.x; /* 256 */
  __shared__ float Tm[4096], Sm[4096];
  const float* st = ws + WS_ST + (size_t)c * 64 * 4096;
  for (int e = tid; e < 4096; e += 256) {
    Tm[e] = st[(32 + kk) * 4096 + e];
    Sm[e] = st[jj * 4096 + e];
  }
  __syncthreads();
  int l = kk * 32 + jj;
  float* qrow = ws + WS_QBIG + (size_t)l * KBIG + c * 4096;
  for (int e = tid; e < 4096; e += 256) {
    int i = e >> 6, j = e & 63; float s = 0.f;
    for (int m = 0; m < 64; ++m) s += Tm[i * 64 + m] * Sm[m * 64 + j];
    qrow[e] = s;
  }
}

/* ============ K4: W[d,(c,m,n)] = C_c[d,m] * Bbar_c[d,n] ==================== */
__global__ void k_buildW(const float* __restrict__ Cp, float* __restrict__ ws) {
  const float* Bb = ws + WS_BBAR;
  float* W = ws + WS_WBIG;
  size_t total = (size_t)D_MODEL * KBIG;
  for (size_t idx = (size_t)blockIdx.x * blockDim.x + threadIdx.x; idx < total;
       idx += (size_t)gridDim.x * blockDim.x) {
    size_t dd = idx / KBIG;
    int rem = (int)(idx % KBIG);
    int c = rem >> 12, m = (rem >> 6) & 63, n = rem & 63;
    W[idx] = Cp[((size_t)c * D_MODEL + dd) * 64 + m] *
             Bb[((size_t)c * D_MODEL + dd) * 64 + n];
  }
}

/* ============ K5: K = W @ Q^T via f32 WMMA =================================
 * All 8 waves of a block share the same M-tile, so the A-strip (16 rows of W)
 * is staged in LDS via a double-buffered global_load_async_to_lds_b128
 * pipeline (ASYNCcnt), cutting A-side L2 traffic 8x.  B streams direct from
 * global with the load/wmma clause pipeline.  LDS rows padded to 260 floats
 * so the 16 row-readers hit distinct banks.                                  */
#define KCH   256                 /* k-chunk (floats) */
#define APAD  260                 /* padded LDS row stride */
#define NCHUNK (KBIG / KCH)       /* 64 */

__global__ void k_gemm_K(const float* __restrict__ Dp, float* __restrict__ ws) {
  const float* W = ws + WS_WBIG;
  const float* Q = ws + WS_QBIG;
  float* KrevT = ws + WS_KREVT;
  int tid = threadIdx.x;                 /* 256 = 8 waves */
  int waveid = tid >> 5, lane = tid & 31;
  int tM = blockIdx.x >> 3;              /* 0..31 : shared by whole block */
  int tN = (blockIdx.x & 7) * 8 + waveid;/* 0..63 : one N-tile per wave   */
  int half = lane >> 4, l15 = lane & 15;
  int koff = half * 2;

  __shared__ float As[2][16 * APAD];
  const float* Abase = W + (size_t)(tM * 16) * KBIG;

  /* issue async copy of chunk ch into buffer b: 16 rows x 64 float4 */
#define LOAD_A_CHUNK(ch, b)                                                   \
  {                                                                           \
    for (int r4 = 0; r4 < 4; ++r4) {                                          \
      int idx = tid + r4 * 256;        /* 0..1023 float4 slots */             \
      int row = idx >> 6;              /* /64 */                              \
      int c4  = idx & 63;                                                     \
      const float* gp = Abase + (size_t)row * KBIG + (ch) * KCH + c4 * 4;     \
      unsigned lp = (unsigned)(size_t)&As[(b)][row * APAD + c4 * 4];          \
      asm volatile("global_load_async_to_lds_b128 %0, %1, off"                \
                   :: "v"(lp), "v"((unsigned long long)(size_t)gp)            \
                   : "memory");                                               \
    }                                                                         \
  }

  v8f acc = {0.f, 0.f, 0.f, 0.f, 0.f, 0.f, 0.f, 0.f};
  LOAD_A_CHUNK(0, 0);
  asm volatile("s_wait_asynccnt 0" ::: "memory");
  __syncthreads();

  const float* Bq = Q + (size_t)(tN * 16 + l15) * KBIG + koff;
  for (int ch = 0; ch < NCHUNK; ++ch) {
    int b = ch & 1;
    if (ch + 1 < NCHUNK) LOAD_A_CHUNK(ch + 1, (ch + 1) & 1);
    const float* Alds = &As[b][l15 * APAD + koff];
    const float* Bg = Bq + ch * KCH;
    for (int k0 = 0; k0 < KCH; k0 += 4) {
      v2f a = *(const v2f*)(Alds + k0);
      v2f bv = *(const v2f*)(Bg + k0);
      acc = WMMA_F32(a, bv, acc);
    }
    asm volatile("s_wait_asynccnt 0" ::: "memory");
    __syncthreads();
  }
#undef LOAD_A_CHUNK

  for (int r = 0; r < 8; ++r) {
    int dd = tM * 16 + r + half * 8;
    int l = tN * 16 + l15;
    float v = acc[r];
    if (l == 0) { /* D skip lands at lag L-1 after the XLA no-flip conv */
      float ds = 0.f;
      for (int c = 0; c < NC; ++c) ds += Dp[c * D_MODEL + dd];
      v += ds;
    }
    KrevT[(size_t)(SEQ_L - 1 - l) * D_MODEL + dd] = v;
  }
}

/* ============ K6: depthwise causal conv, register sliding window =========== */
__global__ void k_conv(const float* __restrict__ x, float* __restrict__ ws) {
  int b = blockIdx.x, cg = blockIdx.y, tc = blockIdx.z;
  int c = cg * 64 + threadIdx.x;  /* blockDim 64 */
  int t0 = tc * 16;
  const float* KrevT = ws + WS_KREVT;
  const float* xb = x + (size_t)b * SEQ_L * D_MODEL;
  float* yb = ws + WS_Y + (size_t)b * SEQ_L * D_MODEL;
  float w[16], acc[16];
#pragma unroll
  for (int i = 0; i < 16; ++i) { acc[i] = 0.f; w[i] = xb[(size_t)(t0 + i) * D_MODEL + c]; }
  int gmax = t0 + 16;
  for (int g = 0; g < gmax; ++g) {
    if ((g & 63) == 0)
      __builtin_prefetch(KrevT + (size_t)(g + 64) * D_MODEL + c, 0, 1);
    float kv = KrevT[(size_t)g * D_MODEL + c];
#pragma unroll
    for (int i = 0; i < 16; ++i) acc[i] += w[i] * kv;
#pragma unroll
    for (int i = 15; i > 0; --i) w[i] = w[i - 1];
    int tin = t0 - 1 - g;
    w[0] = (tin >= 0) ? xb[(size_t)tin * D_MODEL + c] : 0.f;
  }
#pragma unroll
  for (int i = 0; i < 16; ++i) yb[(size_t)(t0 + i) * D_MODEL + c] = acc[i];
}

/* ============ K7a: squeeze (mean over time) ================================ */
__global__ void k_semean(float* __restrict__ ws) {
  int b = blockIdx.x;
  int c = blockIdx.y * 128 + threadIdx.x; /* block 128, grid.y 4 */
  const float* yb = ws + WS_Y + (size_t)b * SEQ_L * D_MODEL;
  float s = 0.f;
  for (int t = 0; t < SEQ_L; ++t) s += yb[(size_t)t * D_MODEL + c];
  ws[WS_SMEAN + b * D_MODEL + c] = s * (1.f / SEQ_L);
}

/* ============ K7b: SE excitation MLP ======================================= */
__global__ void k_segate(const float* __restrict__ w1, const float* __restrict__ b1,
                         const float* __restrict__ w2, const float* __restrict__ b2,
                         float* __restrict__ ws) {
  int b = blockIdx.x, tid = threadIdx.x; /* 256 */
  __shared__ float sh_s[512]; __shared__ float sh_h[64];
  for (int e = tid; e < 512; e += 256) sh_s[e] = ws[WS_SMEAN + b * 512 + e];
  __syncthreads();
  if (tid < 64) {
    float a = b1[tid];
    for (int c = 0; c < 512; ++c) a += w1[tid * 512 + c] * sh_s[c];
    sh_h[tid] = fmaxf(a, 0.f);
  }
  __syncthreads();
  for (int cc = tid; cc < 512; cc += 256) {
    float a = b2[cc];
    for (int r = 0; r < 64; ++r) a += w2[cc * 64 + r] * sh_h[r];
    ws[WS_SGATE + b * 512 + cc] = 1.f / (1.f + expf(-a));
  }
}

/* ============ K8: y1 = LN(x + y*gate) ====================================== */
__global__ void k_y1(const float* __restrict__ x, const float* __restrict__ n1g,
                     const float* __restrict__ n1b, float* __restrict__ ws) {
  int rb = blockIdx.x;            /* 8192 rows */
  int b = rb >> 10;
  int tid = threadIdx.x;          /* 256, 2 elems each */
  const float* xr = x + (size_t)rb * 512;
  const float* yr = ws + WS_Y + (size_t)rb * 512;
  const float* gr = ws + WS_SGATE + b * 512;
  float* o = ws + WS_Y1 + (size_t)rb * 512;
  __shared__ float red[256];
  float v0 = xr[tid] + yr[tid] * gr[tid];
  float v1 = xr[tid + 256] + yr[tid + 256] * gr[tid + 256];
  red[tid] = v0 + v1; __syncthreads();
  for (int s = 128; s > 0; s >>= 1) { if (tid < s) red[tid] += red[tid + s]; __syncthreads(); }
  float mean = red[0] * (1.f / 512.f); __syncthreads();
  float d0 = v0 - mean, d1 = v1 - mean;
  red[tid] = d0 * d0 + d1 * d1; __syncthreads();
  for (int s = 128; s > 0; s >>= 1) { if (tid < s) red[tid] += red[tid + s]; __syncthreads(); }
  float rs = rsqrtf(red[0] * (1.f / 512.f) + EPS);
  o[tid] = d0 * rs * n1g[tid] + n1b[tid];
  o[tid + 256] = d1 * rs * n1g[tid + 256] + n1b[tid + 256];
}

/* ============ K9: up GEMM + fused GLU (gelu * sigmoid) via WMMA =========== */
__device__ __forceinline__ float gelu_exact(float a) {
  return 0.5f * a * (1.f + erff(a * 0.70710678118654752f));
}
__global__ void k_upglu(const float* __restrict__ upw, const float* __restrict__ upb,
                        float* __restrict__ ws) {
  const float* Y1 = ws + WS_Y1;
  float* H = ws + WS_H;
  int wave = (blockIdx.x * blockDim.x + threadIdx.x) >> 5;
  int lane = threadIdx.x & 31;
  int tM = wave >> 6;   /* 0..511 (8192/16) */
  int tN = wave & 63;   /* 0..63  (1024/16) */
  int half = lane >> 4, l15 = lane & 15;
  int koff = half * 2;
  const float* Ap = Y1 + (size_t)(tM * 16 + l15) * 512 + koff;
  const float* Ba = upw + (size_t)(tN * 16 + l15) * 512 + koff;
  const float* Bg = upw + (size_t)(tN * 16 + l15 + 1024) * 512 + koff;
  v8f accA = {0.f, 0.f, 0.f, 0.f, 0.f, 0.f, 0.f, 0.f};
  v8f accG = {0.f, 0.f, 0.f, 0.f, 0.f, 0.f, 0.f, 0.f};
  for (int k0 = 0; k0 < 512; k0 += 4) {
    v2f a = *(const v2f*)(Ap + k0);
    v2f ba = *(const v2f*)(Ba + k0);
    v2f bg = *(const v2f*)(Bg + k0);
    accA = WMMA_F32(a, ba, accA);
    accG = WMMA_F32(a, bg, accG);
  }
  int col = tN * 16 + l15;
  float biasA = upb[col], biasG = upb[col + 1024];
  for (int r = 0; r < 8; ++r) {
    int row = tM * 16 + r + half * 8;
    float av = accA[r] + biasA;
    float gv = accG[r] + biasG;
    H[(size_t)row * HIDDEN + col] = gelu_exact(av) * (1.f / (1.f + expf(-gv)));
  }
}

/* ============ K10: down GEMM via WMMA ====================================== */
__global__ void k_down(const float* __restrict__ dw, const float* __restrict__ db,
                       float* __restrict__ ws) {
  const float* H = ws + WS_H;
  float* Z = ws + WS_ZR;
  int wave = (blockIdx.x * blockDim.x + threadIdx.x) >> 5;
  int lane = threadIdx.x & 31;
  int tM = wave >> 5;   /* 0..511 (8192/16) */
  int tN = wave & 31;   /* 0..31  (512/16)  */
  int half = lane >> 4, l15 = lane & 15;
  int koff = half * 2;
  const float* Ap = H + (size_t)(tM * 16 + l15) * HIDDEN + koff;
  const float* Bp = dw + (size_t)(tN * 16 + l15) * HIDDEN + koff;
  v8f acc = {0.f, 0.f, 0.f, 0.f, 0.f, 0.f, 0.f, 0.f};
  for (int k0 = 0; k0 < HIDDEN; k0 += 4) {
    v2f a = *(const v2f*)(Ap + k0);
    v2f b = *(const v2f*)(Bp + k0);
    acc = WMMA_F32(a, b, acc);
  }
  int col = tN * 16 + l15;
  float bias = db[col];
  for (int r = 0; r < 8; ++r) {
    int row = tM * 16 + r + half * 8;
    Z[(size_t)row * D_MODEL + col] = acc[r] + bias;
  }
}

/* ============ K11: z = LN(y1+zr); out = LN(y1+z) =========================== */
__global__ void k_final(const float* __restrict__ mixg, const float* __restrict__ mixb,
                        const float* __restrict__ n2g, const float* __restrict__ n2b,
                        float* __restrict__ out, float* __restrict__ ws) {
  int rb = blockIdx.x, tid = threadIdx.x; /* 8192 x 256 */
  const float* y1 = ws + WS_Y1 + (size_t)rb * 512;
  const float* zr = ws + WS_ZR + (size_t)rb * 512;
  __shared__ float red[256];
  float a0 = y1[tid] + zr[tid];
  float a1 = y1[tid + 256] + zr[tid + 256];
  red[tid] = a0 + a1; __syncthreads();
  for (int s = 128; s > 0; s >>= 1) { if (tid < s) red[tid] += red[tid + s]; __syncthreads(); }
  float m1 = red[0] * (1.f / 512.f); __syncthreads();
  float d0 = a0 - m1, d1 = a1 - m1;
  red[tid] = d0 * d0 + d1 * d1; __syncthreads();
  for (int s = 128; s > 0; s >>= 1) { if (tid < s) red[tid] += red[tid + s]; __syncthreads(); }
  float rs1 = rsqrtf(red[0] * (1.f / 512.f) + EPS);
  float z0 = d0 * rs1 * mixg[tid] + mixb[tid];
  float z1 = d1 * rs1 * mixg[tid + 256] + mixb[tid + 256];
  __syncthreads();
  float t0v = y1[tid] + z0, t1v = y1[tid + 256] + z1;
  red[tid] = t0v + t1v; __syncthreads();
  for (int s = 128; s > 0; s >>= 1) { if (tid < s) red[tid] += red[tid + s]; __syncthreads(); }
  float m2 = red[0] * (1.f / 512.f); __syncthreads();
  float e0 = t0v - m2, e1 = t1v - m2;
  red[tid] = e0 * e0 + e1 * e1; __syncthreads();
  for (int s = 128; s > 0; s >>= 1) { if (tid < s) red[tid] += red[tid + s]; __syncthreads(); }
  float rs2 = rsqrtf(red[0] * (1.f / 512.f) + EPS);
  float* o = out + (size_t)rb * 512;
  o[tid] = e0 * rs2 * n2g[tid] + n2b[tid];
  o[tid + 256] = e1 * rs2 * n2g[tid + 256] + n2b[tid + 256];
}

/* ============================ launcher ===================================== */
extern "C" void kernel_launch(void* const* d_in, const int* in_sizes, int n_in,
                              void* d_out, int out_size, void* d_ws, size_t ws_size,
                              hipStream_t stream) {
  (void)in_sizes; (void)n_in; (void)out_size; (void)ws_size;
  const float* x     = (const float*)d_in[0];
  const float* Bp    = (const float*)d_in[1];
  const float* Cp    = (const float*)d_in[2];
  const float* Dp    = (const float*)d_in[3];
  const float* ldt   = (const float*)d_in[4];
  const float* se_w1 = (const float*)d_in[5];
  const float* se_b1 = (const float*)d_in[6];
  const float* se_w2 = (const float*)d_in[7];
  const float* se_b2 = (const float*)d_in[8];
  const float* n1_g  = (const float*)d_in[9];
  const float* n1_b  = (const float*)d_in[10];
  const float* up_w  = (const float*)d_in[11];
  const float* up_b  = (const float*)d_in[12];
  const float* dn_w  = (const float*)d_in[13];
  const float* dn_b  = (const float*)d_in[14];
  const float* mix_g = (const float*)d_in[15];
  const float* mix_b = (const float*)d_in[16];
  const float* n2_g  = (const float*)d_in[17];
  const float* n2_b  = (const float*)d_in[18];
  float* out = (float*)d_out;
  float* ws  = (float*)d_ws;

  /* phase 1: SSM kernel generation */
  k_discretize<<<dim3(NC, 9), 64, 0, stream>>>(Bp, ldt, ws);
  k_powers<<<NC, 256, 0, stream>>>(ws);
  k_buildQ<<<dim3(32, 32, NC), 256, 0, stream>>>(ws);
  k_buildW<<<4096, 256, 0, stream>>>(Cp, ws);
  k_gemm_K<<<256, 256, 0, stream>>>(Dp, ws);   /* 2048 waves, 1 tile/wave */

  /* phase 2: conv + SE + LN + ChannelMix */
  k_conv<<<dim3(BATCH, 8, 64), 64, 0, stream>>>(x, ws);
  k_semean<<<dim3(BATCH, 4), 128, 0, stream>>>(ws);
  k_segate<<<BATCH, 256, 0, stream>>>(se_w1, se_b1, se_w2, se_b2, ws);
  k_y1<<<BATCH * SEQ_L, 256, 0, stream>>>(x, n1_g, n1_b, ws);
  k_upglu<<<4096, 256, 0, stream>>>(up_w, up_b, ws);   /* 32768 tiles */
  k_down<<<2048, 256, 0, stream>>>(dn_w, dn_b, ws);    /* 16384 tiles */
  k_final<<<BATCH * SEQ_L, 256, 0, stream>>>(mix_g, mix_b, n2_g, n2_b, out, ws);
}